// ALiBiPositionalBias_22359599743545
// MI455X (gfx1250) — compile-verified
//
#include <hip/hip_runtime.h>

typedef float v2f __attribute__((ext_vector_type(2)));
typedef float v4f __attribute__((ext_vector_type(4)));
typedef float v8f __attribute__((ext_vector_type(8)));

// Reference module constants (fixed in the reference file).
constexpr int B_ = 2, H_ = 16, S_ = 2048;
constexpr int TS    = 16;        // WMMA tile edge
constexpr int TJ    = S_ / TS;   // 128 tile-columns (7 bits)
constexpr int WAVES = 8;         // waves per 256-thread block (wave32)

__global__ __launch_bounds__(256)
void alibi_wmma_kernel(const float* __restrict__ scores,
                       const float* __restrict__ slope_adj,
                       const float* __restrict__ slope_scal,
                       const float* __restrict__ mscale,
                       const float* __restrict__ seq_bias,
                       const float* __restrict__ hier_bias,
                       float* __restrict__ out)
{
    const int wave = threadIdx.x >> 5;              // wave32
    const int lane = threadIdx.x & 31;
    const unsigned t = blockIdx.x * WAVES + wave;   // one 16x16 tile per wave

    // tile id bits: [b:1][h:4][it:7][jt:7]
    const int jt = t & (TJ - 1);
    const int it = (t >> 7) & (TJ - 1);
    const int h  = (t >> 14) & (H_ - 1);
    const int b  = t >> 18;

    const int i0 = it * TS, j0 = jt * TS;
    const size_t base = ((size_t)(b * H_ + h) * S_ + i0) * S_ + j0;
    float* obase = out + base;

    const int half = lane >> 4;   // 0: lanes 0-15, 1: lanes 16-31
    const int n    = lane & 15;   // column within tile

    const float NEG_INF = -__builtin_huge_valf();

    // ---- Structurally masked tile: all j > i. Output is -inf independent of
    // scores -> skip the 1 KiB load entirely (~50% of tiles, halves read BW).
    if (j0 > i0 + (TS - 1)) {
        const int r  = lane >> 1;            // row 0..15 (2 lanes per row)
        const int c4 = (lane & 1) * 8;       // each lane covers 8 floats
        v4f m = { NEG_INF, NEG_INF, NEG_INF, NEG_INF };
        __builtin_nontemporal_store(m, (v4f*)(obase + (size_t)r * S_ + c4));
        __builtin_nontemporal_store(m, (v4f*)(obase + (size_t)r * S_ + c4 + 4));
        return;
    }

    // ---- Per-head scalars (uniform within the wave).
    // base slope for power-of-two H: 2^(-8*(h+1)/H)
    const float base_slope = exp2f(-8.0f * (float)(h + 1) / (float)H_);
    const float slope = (base_slope * slope_scal[h] + slope_adj[h]) * mscale[h];
    const float seq   = seq_bias[h];
    const float hier  = hier_bias[h];

    const float* tbase = scores + base;

    // ---- Load score tile directly in the WMMA 16x16 f32 C/D layout:
    // VGPR v <-> row (v + 8*half), lane <-> column n. Non-temporal (streamed once).
    v8f c;
    #pragma unroll
    for (int v = 0; v < 8; ++v)
        c[v] = __builtin_nontemporal_load(tbase + (size_t)(v + 8 * half) * S_ + n);

    // ---- Rank-2 factorization of slope*(i-j), fused with the score add:
    //   A[m,:] = { slope*(i0+m), 1, 0, 0 }   (16x4)
    //   B[:,n] = { 1, -slope*(j0+n), 0, 0 }  (4x16)
    //   D = A x B + C  ->  scores + slope*(i-j)  on the matrix pipe.
    // A layout: lanes 0-15 rows M=0..15; VGPR0 = K0 (K2 in upper lanes), VGPR1 = K1 (K3).
    v2f a, bm;
    a.x  = half ? 0.0f : slope * (float)(i0 + lane);
    a.y  = half ? 0.0f : 1.0f;
    bm.x = half ? 0.0f : 1.0f;
    bm.y = half ? 0.0f : (-slope * (float)(j0 + n));

    v8f d = __builtin_amdgcn_wmma_f32_16x16x4_f32(
        /*neg_a=*/false, a, /*neg_b=*/false, bm,
        /*c_mod=*/(short)0, c, /*reuse_a=*/false, /*reuse_b=*/false);

    // ---- Branchless non-separable fixup: causal step, hierarchical pattern,
    // -inf mask. The three pattern sets are disjoint, so cndmask chain order
    // is irrelevant. d = i - j increments by 1 per v, so the %10 residue is
    // computed ONCE per lane and tested per-v against an unroll-time constant.
    const int dd0 = (i0 + 8 * half) - (j0 + n);          // d at v = 0 (>= -30 here)
    const unsigned r0 = (unsigned)(dd0 + 40) % 10u;      // residue of d mod 10

    #pragma unroll
    for (int v = 0; v < 8; ++v) {
        const int dd = dd0 + v;
        const unsigned u = (unsigned)dd;

        // fib {1,2,3,5,8,13,21}: bitmask test, bitwise (no short-circuit)
        const bool fib = (u < 32u) & (((0x20212Eu >> (u & 31u)) & 1u) != 0u);
        // pow {4,16,64,256}
        const bool pw  = (u == 4u) | (u == 16u) | (u == 64u) | (u == 256u);
        // d % 10 == 0  <=>  r0 == (10 - v) % 10   (single compare, const RHS)
        const bool m10 = (r0 == ((10u - (unsigned)v) % 10u));

        float pat = m10 ? 0.02f : 0.0f;
        pat       = pw  ? 0.05f : pat;
        pat       = fib ? 0.10f : pat;

        float val = d[v] + __builtin_fmaf(pat, hier, seq);  // d = score + slope*(i-j)
        val = (dd < 0) ? NEG_INF : val;                     // strictly upper triangle

        __builtin_nontemporal_store(val, obase + (size_t)(v + 8 * half) * S_ + n);
    }
}

extern "C" void kernel_launch(void* const* d_in, const int* in_sizes, int n_in,
                              void* d_out, int out_size, void* d_ws, size_t ws_size,
                              hipStream_t stream)
{
    (void)in_sizes; (void)n_in; (void)out_size; (void)d_ws; (void)ws_size;

    const float* scores     = (const float*)d_in[0];
    const float* slope_adj  = (const float*)d_in[1];
    const float* slope_scal = (const float*)d_in[2];
    const float* mscale     = (const float*)d_in[3];
    const float* seq_bias   = (const float*)d_in[4];
    const float* hier_bias  = (const float*)d_in[5];
    float* out              = (float*)d_out;

    const int nTiles = B_ * H_ * TJ * TJ;         // 524288 tiles, 1 per wave
    const int grid   = nTiles / WAVES;            // 65536 blocks of 8 waves

    alibi_wmma_kernel<<<grid, 256, 0, stream>>>(
        scores, slope_adj, slope_scal, mscale, seq_bias, hier_bias, out);
}